// MixedModel_23957327577306
// MI455X (gfx1250) — compile-verified
//
#include <hip/hip_runtime.h>
#include <math.h>

// CDNA5 / gfx1250, wave32. FP32 WMMA 16x16x4 for all GEMMs.

typedef float v2f __attribute__((ext_vector_type(2)));
typedef float v8f __attribute__((ext_vector_type(8)));

#define F_IN 128
#define H    128
#define H2   256
#define H3   384

// ---------------- float atomic max (two-sided int trick) ----------------
__device__ inline void atomicMaxF32(float* addr, float val) {
  if (val >= 0.0f) {
    atomicMax((int*)addr, __float_as_int(val));
  } else {
    atomicMin((unsigned int*)addr, (unsigned int)__float_as_int(val));
  }
}

// ---------------- WMMA fp32 GEMM ----------------
// D[M,Ncols] = act(A[M,K]) @ B[K,Ncols] (+bias) (+=D if accumulate)
// amode: 0 = identity, 1 = relu, 2 = BN(mu,rstd,gamma,beta per A-col)+relu
#define BM 128
#define BN 64
#define BK 32
#define LDA_S 36    // padded LDS stride for A tile rows (conflict-free b64 frags)
#define LDB_S 160   // K-pair row stride for B tile (mod 64 == 32 -> half-waves disjoint)

__global__ __launch_bounds__(256)
void gemm_wmma_f32(const float* __restrict__ A, int lda, int M, int K,
                   const float* __restrict__ B, int ldb, int Ncols,
                   float* __restrict__ D, int ldd,
                   const float* __restrict__ bias, int accumulate, int amode,
                   const float* __restrict__ mu, const float* __restrict__ rstd,
                   const float* __restrict__ gamma, const float* __restrict__ beta)
{
  __shared__ float As[BM * LDA_S];
  // B tile, K-paired layout: (k,n) -> Bs[(k>>1)*LDB_S + n*2 + (k&1)]
  __shared__ float Bs[(BK / 2) * LDB_S];

  const int tid  = threadIdx.x;
  const int mblk = blockIdx.x * BM;
  const int nblk = blockIdx.y * BN;
  const int w    = tid >> 5;        // wave id 0..7
  const int lane = tid & 31;
  const int half = lane >> 4;       // 0: K frag {0,1}, 1: K frag {2,3}
  const int lr   = lane & 15;
  const int wrow = w * 16;          // 16 output rows per wave

  v8f acc[4];
  #pragma unroll
  for (int t = 0; t < 4; ++t) acc[t] = (v8f){};

  // cooperative-load coordinates
  const int ar  = tid >> 1;         // A tile row 0..127
  const int ac  = (tid & 1) * 16;   // A tile col base 0/16 (16 floats each)
  const int bp  = tid >> 4;         // B pair index 0..15 (K rows 2bp, 2bp+1)
  const int bc4 = (tid & 15) * 4;   // B col base (4 cols each)

  for (int k0 = 0; k0 < K; k0 += BK) {
    // ---- stage A tile (activation fused) ----
    {
      const int grow = mblk + ar;
      float* dst = &As[ar * LDA_S + ac];
      if (grow < M) {
        const float* src = &A[(long)grow * lda + k0 + ac];
        #pragma unroll
        for (int j = 0; j < 16; ++j) {
          float v = src[j];
          if (amode == 2) {
            const int col = k0 + ac + j;
            v = (v - mu[col]) * rstd[col] * gamma[col] + beta[col];
            v = fmaxf(v, 0.0f);
          } else if (amode == 1) {
            v = fmaxf(v, 0.0f);
          }
          dst[j] = v;
        }
      } else {
        #pragma unroll
        for (int j = 0; j < 16; ++j) dst[j] = 0.0f;
      }
    }
    // ---- stage B tile in K-paired layout (zero-pad past Ncols) ----
    {
      const int kr = k0 + 2 * bp;
      float* dst = &Bs[bp * LDB_S + bc4 * 2];
      #pragma unroll
      for (int j = 0; j < 4; ++j) {
        const int gc = nblk + bc4 + j;
        float v0 = 0.0f, v1 = 0.0f;
        if (gc < Ncols) {
          v0 = B[(long)kr * ldb + gc];
          v1 = B[(long)(kr + 1) * ldb + gc];
        }
        dst[j * 2]     = v0;
        dst[j * 2 + 1] = v1;
      }
    }
    __syncthreads();

    // ---- WMMA over this K chunk ----
    #pragma unroll
    for (int kk = 0; kk < BK; kk += 4) {
      // A frag: rows M = wrow+lr, K = kk+2*half, kk+2*half+1 (adjacent -> b64)
      const v2f a = *(const v2f*)&As[(wrow + lr) * LDA_S + kk + 2 * half];
      const int p = (kk >> 1) + half;   // B pair row for this half-wave
      #pragma unroll
      for (int t = 0; t < 4; ++t) {
        const v2f b = *(const v2f*)&Bs[p * LDB_S + (t * 16 + lr) * 2];
        acc[t] = __builtin_amdgcn_wmma_f32_16x16x4_f32(
            false, a, false, b, (short)0, acc[t], false, false);
      }
    }
    __syncthreads();
  }

  // ---- store D (C/D layout: vgpr g -> M=g (+8 for upper half), N=lane) ----
  #pragma unroll
  for (int t = 0; t < 4; ++t) {
    #pragma unroll
    for (int g = 0; g < 8; ++g) {
      const int row = mblk + wrow + g + 8 * half;
      const int col = nblk + t * 16 + lr;
      if (row < M && col < Ncols) {
        float v = acc[t][g];
        if (bias) v += bias[col];
        const long o = (long)row * ldd + col;
        if (accumulate) D[o] += v; else D[o] = v;
      }
    }
  }
}

// ---------------- small helper kernels ----------------
__global__ void init_node_kernel(float* m, float* denom, float* cnt,
                                 float* bnsum, float* bnsq, int N) {
  const int i = blockIdx.x * blockDim.x + threadIdx.x;
  if (i < N) { m[i] = -INFINITY; denom[i] = 0.0f; cnt[i] = 0.0f; }
  if (i < H3) { bnsum[i] = 0.0f; bnsq[i] = 0.0f; }
}

__global__ void init_hcat_kernel(float* hcat, const float* b_gat,
                                 const float* b_gcn, long total) {
  const long i = blockIdx.x * (long)blockDim.x + threadIdx.x;
  if (i >= total) return;
  const int c = (int)(i % H3);
  float v = 0.0f;
  if (c < H) v = b_gat[c];
  else if (c < 2 * H) v = b_gcn[c - H];
  hcat[i] = v;   // SAGE columns start at 0 (bias added by GEMM)
}

__global__ void zero_kernel(float* p, long n) {
  const long i = blockIdx.x * (long)blockDim.x + threadIdx.x;
  if (i < n) p[i] = 0.0f;
}

// per-node dots with attention vectors: one wave32 per node (coalesced row)
__global__ void attn_dots_kernel(const float* __restrict__ hg,
                                 const float* __restrict__ att_s,
                                 const float* __restrict__ att_d,
                                 float* a_s, float* a_d, int N) {
  const int wave = (int)((blockIdx.x * (long)blockDim.x + threadIdx.x) >> 5);
  const int lane = threadIdx.x & 31;
  if (wave >= N) return;
  const float* row = hg + (long)wave * H;
  float s1 = 0.0f, s2 = 0.0f;
  for (int j = lane; j < H; j += 32) {
    const float v = row[j];
    s1 += v * att_s[j];
    s2 += v * att_d[j];
  }
  for (int off = 16; off > 0; off >>= 1) {
    s1 += __shfl_xor(s1, off, 32);
    s2 += __shfl_xor(s2, off, 32);
  }
  if (lane == 0) { a_s[wave] = s1; a_d[wave] = s2; }
}

// pass A: leaky-relu edge scores -> segment max; also in-degree (no self loop)
__global__ void edge_pass_a(const int* __restrict__ ei, int E, int N,
                            const float* __restrict__ a_s,
                            const float* __restrict__ a_d,
                            float* m, float* cnt) {
  const int i = blockIdx.x * blockDim.x + threadIdx.x;
  const int EE = E + N;
  if (i >= EE) return;
  int s, d;
  if (i < E) { s = ei[i]; d = ei[E + i]; atomicAdd(&cnt[d], 1.0f); }
  else       { s = d = i - E; }                 // self loop
  const float t = a_s[s] + a_d[d];
  const float e = (t > 0.0f) ? t : 0.2f * t;
  atomicMaxF32(&m[d], e);
}

// pass B: softmax numerator + segment sum
__global__ void edge_pass_b(const int* __restrict__ ei, int E, int N,
                            const float* __restrict__ a_s,
                            const float* __restrict__ a_d,
                            const float* __restrict__ m,
                            float* ex, float* denom) {
  const int i = blockIdx.x * blockDim.x + threadIdx.x;
  const int EE = E + N;
  if (i >= EE) return;
  int s, d;
  if (i < E) { s = ei[i]; d = ei[E + i]; }
  else       { s = d = i - E; }
  const float t = a_s[s] + a_d[d];
  const float e = (t > 0.0f) ? t : 0.2f * t;
  const float v = __expf(e - m[d]);
  ex[i] = v;
  atomicAdd(&denom[d], v);
}

// fused GAT+GCN feature scatter: one wave32 = one edge (float4 per lane,
// 512B coalesced gathers, per-edge scalars loaded once per wave)
__global__ void scatter_gat_gcn(const int* __restrict__ ei, int E, int N,
                                const float* __restrict__ ex,
                                const float* __restrict__ denom,
                                const float* __restrict__ cnt,
                                const float* __restrict__ hg,
                                const float* __restrict__ hc,
                                float* hcat) {
  const long idx = blockIdx.x * (long)blockDim.x + threadIdx.x;
  const long EE = (long)E + N;
  if (idx >= EE * 32) return;
  const int edge = (int)(idx >> 5);
  const int f    = (int)(idx & 31) * 4;
  int s, d;
  if (edge < E) { s = ei[edge]; d = ei[E + edge]; }
  else          { s = d = edge - E; }
  const float alpha = ex[edge] / denom[d];
  const float norm  = rsqrtf(cnt[s] + 1.0f) * rsqrtf(cnt[d] + 1.0f);
  const float4 g = *(const float4*)&hg[(long)s * H + f];
  const float4 c = *(const float4*)&hc[(long)s * H + f];
  float* pg = &hcat[(long)d * H3 + f];
  float* pc = &hcat[(long)d * H3 + H + f];
  atomicAdd(pg + 0, alpha * g.x);
  atomicAdd(pg + 1, alpha * g.y);
  atomicAdd(pg + 2, alpha * g.z);
  atomicAdd(pg + 3, alpha * g.w);
  atomicAdd(pc + 0, norm * c.x);
  atomicAdd(pc + 1, norm * c.y);
  atomicAdd(pc + 2, norm * c.z);
  atomicAdd(pc + 3, norm * c.w);
}

// SAGE neighbor sum (no self loops), float4 granule
__global__ void scatter_sage(const int* __restrict__ ei, int E,
                             const float* __restrict__ x, float* agg) {
  const long idx = blockIdx.x * (long)blockDim.x + threadIdx.x;
  if (idx >= (long)E * 32) return;
  const int edge = (int)(idx >> 5);
  const int f    = (int)(idx & 31) * 4;
  const int s = ei[edge], d = ei[E + edge];
  const float4 v = *(const float4*)&x[(long)s * H + f];
  float* p = &agg[(long)d * H + f];
  atomicAdd(p + 0, v.x);
  atomicAdd(p + 1, v.y);
  atomicAdd(p + 2, v.z);
  atomicAdd(p + 3, v.w);
}

__global__ void sage_mean_kernel(float* agg, const float* cnt, long total4) {
  const long idx = blockIdx.x * (long)blockDim.x + threadIdx.x;
  if (idx >= total4) return;                 // total4 = N*32 (float4 granules)
  const int node = (int)(idx >> 5);
  float c = cnt[node];
  c = (c > 1.0f) ? c : 1.0f;
  const float r = 1.0f / c;
  float4* p = (float4*)agg + idx;
  float4 v = *p;
  v.x *= r; v.y *= r; v.z *= r; v.w *= r;
  *p = v;
}

// batchnorm column sums: 384 threads per block (col = tid), 128-row tiles
__global__ __launch_bounds__(384)
void bn_stats_kernel(const float* __restrict__ hcat, float* bnsum, float* bnsq, int N) {
  const int t  = threadIdx.x;       // column 0..383
  const int r0 = blockIdx.x * 128;
  float s = 0.0f, ss = 0.0f;
  for (int r = 0; r < 128; ++r) {
    const int row = r0 + r;
    if (row < N) {
      const float v = hcat[(long)row * H3 + t];
      s += v; ss += v * v;
    }
  }
  atomicAdd(&bnsum[t], s);
  atomicAdd(&bnsq[t], ss);
}

__global__ void bn_final_kernel(const float* bnsum, const float* bnsq,
                                float* mu, float* rstd, int N) {
  const int t = threadIdx.x;
  if (t >= H3) return;
  const float m = bnsum[t] / (float)N;
  const float v = bnsq[t] / (float)N - m * m;
  mu[t] = m;
  rstd[t] = rsqrtf(v + 1e-5f);
}

// ---------------- launcher ----------------
static inline int ceil_div_i(long a, long b) { return (int)((a + b - 1) / b); }

extern "C" void kernel_launch(void* const* d_in, const int* in_sizes, int n_in,
                              void* d_out, int out_size, void* d_ws, size_t ws_size,
                              hipStream_t stream) {
  const float* x        = (const float*)d_in[0];
  const int*   ei       = (const int*)d_in[1];
  const float* W_gat    = (const float*)d_in[2];
  const float* att_src  = (const float*)d_in[3];
  const float* att_dst  = (const float*)d_in[4];
  const float* b_gat    = (const float*)d_in[5];
  const float* W_gcn    = (const float*)d_in[6];
  const float* b_gcn    = (const float*)d_in[7];
  const float* W_sage_l = (const float*)d_in[8];
  const float* b_sage_l = (const float*)d_in[9];
  const float* W_sage_r = (const float*)d_in[10];
  const float* W1       = (const float*)d_in[11];
  const float* b1       = (const float*)d_in[12];
  const float* W2       = (const float*)d_in[13];
  const float* b2       = (const float*)d_in[14];
  const float* W3       = (const float*)d_in[15];
  const float* b3       = (const float*)d_in[16];
  const float* gamma    = (const float*)d_in[17];
  const float* beta     = (const float*)d_in[18];
  float* out = (float*)d_out;

  const int N   = in_sizes[0] / F_IN;
  const int E   = in_sizes[1] / 2;
  const int OUT = in_sizes[15] / H;

  // workspace layout (floats); t1/t2 alias dead buffers
  float* ws = (float*)d_ws;
  long off = 0;
  float* hcat  = ws + off;                 off += (long)N * H3;
  const long hg_off = off;
  float* hg    = ws + off;                 off += (long)N * H;   // x@W_gat
  float* hc    = ws + off;                 off += (long)N * H;   // x@W_gcn
  const long hs_off = off;
  float* hs    = ws + off;                 off += (long)N * H;   // sage agg -> mean
  float* a_s   = ws + off;                 off += N;
  float* a_d   = ws + off;                 off += N;
  float* mvec  = ws + off;                 off += N;
  float* denom = ws + off;                 off += N;
  float* cnt   = ws + off;                 off += N;
  float* bnsum = ws + off;                 off += H3;
  float* bnsq  = ws + off;                 off += H3;
  float* muv   = ws + off;                 off += H3;
  float* rstdv = ws + off;                 off += H3;
  float* exv   = ws + off;                 off += (long)(E + N);
  float* t1 = ws + hg_off;   // [N,256] aliases hg+hc (dead after scatters)
  float* t2 = ws + hs_off;   // [N,128] aliases hs   (dead after sage GEMM)
  (void)ws_size; (void)n_in;

  const int TB = 256;
  const int EE = E + N;

  // init
  init_node_kernel<<<ceil_div_i((N > H3 ? N : H3), TB), TB, 0, stream>>>(
      mvec, denom, cnt, bnsum, bnsq, N);
  init_hcat_kernel<<<ceil_div_i((long)N * H3, TB), TB, 0, stream>>>(
      hcat, b_gat, b_gcn, (long)N * H3);
  zero_kernel<<<ceil_div_i((long)N * H, TB), TB, 0, stream>>>(hs, (long)N * H);

  // branch GEMMs: hg = x@W_gat ; hc = x@W_gcn ; hcat[:,256:] = x@W_sage_r + b_sage_l
  dim3 gA(ceil_div_i(N, BM), (H + BN - 1) / BN);
  gemm_wmma_f32<<<gA, TB, 0, stream>>>(x, F_IN, N, F_IN, W_gat, H, H,
      hg, H, nullptr, 0, 0, nullptr, nullptr, nullptr, nullptr);
  gemm_wmma_f32<<<gA, TB, 0, stream>>>(x, F_IN, N, F_IN, W_gcn, H, H,
      hc, H, nullptr, 0, 0, nullptr, nullptr, nullptr, nullptr);
  gemm_wmma_f32<<<gA, TB, 0, stream>>>(x, F_IN, N, F_IN, W_sage_r, H, H,
      hcat + 2 * H, H3, b_sage_l, 0, 0, nullptr, nullptr, nullptr, nullptr);

  // GAT attention scalars and edge passes
  attn_dots_kernel<<<ceil_div_i((long)N * 32, TB), TB, 0, stream>>>(
      hg, att_src, att_dst, a_s, a_d, N);
  edge_pass_a<<<ceil_div_i(EE, TB), TB, 0, stream>>>(ei, E, N, a_s, a_d, mvec, cnt);
  edge_pass_b<<<ceil_div_i(EE, TB), TB, 0, stream>>>(ei, E, N, a_s, a_d, mvec, exv, denom);

  // feature scatters (one wave per edge, float4 per lane)
  scatter_gat_gcn<<<ceil_div_i((long)EE * 32, TB), TB, 0, stream>>>(
      ei, E, N, exv, denom, cnt, hg, hc, hcat);
  scatter_sage<<<ceil_div_i((long)E * 32, TB), TB, 0, stream>>>(ei, E, x, hs);
  sage_mean_kernel<<<ceil_div_i((long)N * 32, TB), TB, 0, stream>>>(
      hs, cnt, (long)N * 32);

  // hcat[:,256:] += mean @ W_sage_l
  gemm_wmma_f32<<<gA, TB, 0, stream>>>(hs, H, N, H, W_sage_l, H, H,
      hcat + 2 * H, H3, nullptr, 1, 0, nullptr, nullptr, nullptr, nullptr);

  // batchnorm stats
  bn_stats_kernel<<<ceil_div_i(N, 128), 384, 0, stream>>>(hcat, bnsum, bnsq, N);
  bn_final_kernel<<<1, H3, 0, stream>>>(bnsum, bnsq, muv, rstdv, N);

  // MLP: t1 = relu(BN(hcat))@W1+b1 ; t2 = relu(t1)@W2+b2 ; out = relu(t2)@W3+b3
  dim3 g1(ceil_div_i(N, BM), (H2 + BN - 1) / BN);
  gemm_wmma_f32<<<g1, TB, 0, stream>>>(hcat, H3, N, H3, W1, H2, H2,
      t1, H2, b1, 0, 2, muv, rstdv, gamma, beta);
  dim3 g2(ceil_div_i(N, BM), (H + BN - 1) / BN);
  gemm_wmma_f32<<<g2, TB, 0, stream>>>(t1, H2, N, H2, W2, H, H,
      t2, H, b2, 0, 1, nullptr, nullptr, nullptr, nullptr);
  dim3 g3(ceil_div_i(N, BM), (OUT + BN - 1) / BN);
  gemm_wmma_f32<<<g3, TB, 0, stream>>>(t2, H, N, H, W3, OUT, OUT,
      out, OUT, b3, 0, 1, nullptr, nullptr, nullptr, nullptr);
}